// SoftTopKRegionSelection_74517682585705
// MI455X (gfx1250) — compile-verified
//
#include <hip/hip_runtime.h>
#include <hip/hip_bf16.h>

// Problem constants from the reference (setup_inputs is fixed):
#define B_    8
#define C_    256
#define HG    128
#define WG    128
#define HU    256
#define WU    256
#define NPIX  (HU * WU)          // 65536
// k = int(0.3 * 65536) = 19660 ; ascending rank index = N - k - 1
#define RANK0 45875
#define TEMP  10.0f
#define MINW  0.1f

typedef __attribute__((ext_vector_type(4))) float v4f;  // native vector: OK for nontemporal builtins

// ---------- order-preserving float <-> uint key ----------
__device__ __forceinline__ unsigned int fkey(float f) {
  unsigned int u = __float_as_uint(f);
  return (u & 0x80000000u) ? ~u : (u | 0x80000000u);
}
__device__ __forceinline__ float keyf(unsigned int k) {
  unsigned int u = (k & 0x80000000u) ? (k & 0x7FFFFFFFu) : ~k;
  return __uint_as_float(u);
}

// Bilinear 2x upsample, align_corners=False: src = o/2 - 0.25, edge clamped.
__device__ __forceinline__ float upsample_at(const float* __restrict__ a, int oy, int ox) {
  int iy0, iy1; float wy0, wy1;
  if (oy & 1) { iy0 = oy >> 1; iy1 = iy0 + 1; if (iy1 > HG - 1) iy1 = HG - 1; wy0 = 0.75f; wy1 = 0.25f; }
  else        { iy1 = oy >> 1; iy0 = iy1 - 1; if (iy0 < 0) iy0 = 0;           wy0 = 0.25f; wy1 = 0.75f; }
  int ix0, ix1; float wx0, wx1;
  if (ox & 1) { ix0 = ox >> 1; ix1 = ix0 + 1; if (ix1 > WG - 1) ix1 = WG - 1; wx0 = 0.75f; wx1 = 0.25f; }
  else        { ix1 = ox >> 1; ix0 = ix1 - 1; if (ix0 < 0) ix0 = 0;           wx0 = 0.25f; wx1 = 0.75f; }
  float r0 = wx0 * a[iy0 * WG + ix0] + wx1 * a[iy0 * WG + ix1];
  float r1 = wx0 * a[iy1 * WG + ix0] + wx1 * a[iy1 * WG + ix1];
  return wy0 * r0 + wy1 * r1;
}

// ---------------------------------------------------------------------------
// Kernel A: one block per batch. Exact kth-smallest via 3-pass radix select
// (12 + 12 + 8 key bits, 4096-bin LDS histogram), then write sigmoid mask.
// ---------------------------------------------------------------------------
__global__ void __launch_bounds__(1024) topk_mask_kernel(
    const float* __restrict__ att, float* __restrict__ mask, int rank0) {
  __shared__ unsigned int hist[4096];
  __shared__ unsigned int part[1024];
  __shared__ unsigned int s_sel;
  __shared__ unsigned int s_rank;
  __shared__ float s_thr;

  const int b = blockIdx.x;
  const float* a = att + (size_t)b * HG * WG;
  float* m = mask + (size_t)b * NPIX;
  const int t = threadIdx.x;
  if (t == 0) { s_rank = (unsigned int)rank0; s_sel = 0u; }

  for (int pass = 0; pass < 3; ++pass) {
    for (int i = t; i < 4096; i += 1024) hist[i] = 0u;
    __syncthreads();
    const unsigned int prefix = s_sel;

    for (int j = 0; j < NPIX / 1024; ++j) {
      int idx = j * 1024 + t;
      unsigned int key = fkey(upsample_at(a, idx >> 8, idx & (WU - 1)));
      if (pass == 0) {
        atomicAdd(&hist[key >> 20], 1u);
      } else if (pass == 1) {
        if ((key >> 20) == prefix) atomicAdd(&hist[(key >> 8) & 0xFFFu], 1u);
      } else {
        if ((key >> 8) == prefix) atomicAdd(&hist[key & 0xFFu], 1u);
      }
    }
    __syncthreads();
    part[t] = hist[4 * t] + hist[4 * t + 1] + hist[4 * t + 2] + hist[4 * t + 3];
    __syncthreads();

    if (t == 0) {
      unsigned int r = s_rank;
      int g = 0;
      while (g < 1023 && r >= part[g]) { r -= part[g]; ++g; }
      int bkt = 4 * g;
      while (bkt < 4 * g + 3 && r >= hist[bkt]) { r -= hist[bkt]; ++bkt; }
      s_rank = r;
      unsigned int sel;
      if (pass == 0)      sel = (unsigned int)bkt;
      else if (pass == 1) sel = (s_sel << 12) | (unsigned int)bkt;
      else                sel = (s_sel << 8)  | (unsigned int)bkt;
      s_sel = sel;
      if (pass == 2) s_thr = keyf(sel);
    }
    __syncthreads();
  }

  const float thr = s_thr;
  for (int j = 0; j < NPIX / 1024; ++j) {
    int idx = j * 1024 + t;
    float v = upsample_at(a, idx >> 8, idx & (WU - 1));
    m[idx] = 1.0f / (1.0f + __expf(-TEMP * (v - thr)));
  }
}

// ---------------------------------------------------------------------------
// Kernel B: one block per (batch, row). Async-load the mask row into LDS
// (CDNA5 global_load_async_to_lds_b32 + s_wait_asynccnt), then stream all 256
// channels with nontemporal b128 loads/stores + global prefetch.
// ---------------------------------------------------------------------------
__global__ void __launch_bounds__(256) weight_apply_kernel(
    const float* __restrict__ feat, const float* __restrict__ mask,
    float* __restrict__ out) {
  __shared__ float srow[WU];
  const int b = blockIdx.x >> 8;    // HU == 256 rows per batch
  const int h = blockIdx.x & 255;
  const int t = threadIdx.x;

  // Per-lane async copy of one mask element into LDS. Flat address of an LDS
  // location carries the LDS byte offset in its low 32 bits (ISA 10.2).
  const float* gsrc = mask + (size_t)b * NPIX + (size_t)h * WU + t;
  unsigned int lds_off = (unsigned int)(unsigned long long)(&srow[t]);
  asm volatile("global_load_async_to_lds_b32 %0, %1, off"
               :: "v"(lds_off), "v"(gsrc) : "memory");
  asm volatile("s_wait_asynccnt 0x0" ::: "memory");
  __syncthreads();

  const int w4 = (t & 63) * 4;      // float4 column within the row
  const int cbase = t >> 6;         // channel sub-lane 0..3
  v4f sc = *(const v4f*)&srow[w4];  // ds_load_b128
  sc += (v4f)(MINW);

  size_t base = (((size_t)(b * C_ + cbase) * HU + h) * WU) + w4;
  const size_t cstride = (size_t)4 * HU * WU; // 4 channels per iteration

#pragma unroll 4
  for (int cg = 0; cg < C_ / 4; ++cg) {
    size_t off = base + (size_t)cg * cstride;
    __builtin_prefetch(feat + off + 2 * cstride, 0, 1);   // global_prefetch_b8
    v4f v = __builtin_nontemporal_load((const v4f*)(feat + off));
    v4f r = v * sc;
    __builtin_nontemporal_store(r, (v4f*)(out + off));
  }
}

// ---------------------------------------------------------------------------
extern "C" void kernel_launch(void* const* d_in, const int* in_sizes, int n_in,
                              void* d_out, int out_size, void* d_ws, size_t ws_size,
                              hipStream_t stream) {
  (void)in_sizes; (void)n_in; (void)d_ws; (void)ws_size; (void)out_size;
  const float* feat = (const float*)d_in[0];   // (8,256,256,256) f32
  const float* att  = (const float*)d_in[1];   // (8,1,128,128)  f32
  float* out_w = (float*)d_out;                              // weighted feat
  float* out_m = (float*)d_out + (size_t)B_ * C_ * HU * WU;  // selection mask

  topk_mask_kernel<<<B_, 1024, 0, stream>>>(att, out_m, RANK0);
  weight_apply_kernel<<<B_ * HU, 256, 0, stream>>>(feat, out_m, out_w);
}